// model_9646496546938
// MI455X (gfx1250) — compile-verified
//
#include <hip/hip_runtime.h>

typedef float v2f __attribute__((ext_vector_type(2)));
typedef float v8f __attribute__((ext_vector_type(8)));

// ---------------- LDS layout (float/dword indices) ----------------
// x0 tile : 6 rows x 66 cols   (rows i0-1..i0+4, col 0 == image col -1)
// tbl     : 192 ints, patch offset (dy*pitch+dx) per k
// x1 tile : 12 rows x 132 cols (rows 2*i0-2.., col 0 == image col -2)
// x2 tile : 25 rows x 264 cols (rows 4*i0-4.., col 0 == image col -4)
//           row 24 stays zero -> serves the K-padding k=189..191 (dy=12)
// Wt      : weight as [k/2][n] float-pairs, row pitch 160 dwords
//           (160 mod 64 == 32 -> b64 B-fragment loads are bank-perfect)
// logits  : 256 x 68 floats, ALIASED over x1/x2/Wt after the k-loop
#define X0_OFF    0
#define TBL_OFF   396
#define X1_OFF    588
#define X2_OFF    2172
#define WT_OFF    8772
#define WT_PITCH  160
#define SMEM_FLOATS (WT_OFF + 96 * WT_PITCH)      // 24132 floats
#define SMEM_BYTES  (SMEM_FLOATS * 4)             // 96528 B (< 320KB/WGP)
#define LOG_OFF   588
#define LOG_PITCH 68

static __device__ __forceinline__ v8f wmma4(v2f a, v2f b, v8f c) {
#if __has_builtin(__builtin_amdgcn_wmma_f32_16x16x4_f32)
  // 8 args: (neg_a, A, neg_b, B, c_mod, C, reuse_a, reuse_b)
  return __builtin_amdgcn_wmma_f32_16x16x4_f32(false, a, false, b,
                                               (short)0, c, false, false);
#else
  v8f d;
  asm volatile("v_wmma_f32_16x16x4_f32 %0, %1, %2, %3"
               : "=v"(d) : "v"(a), "v"(b), "v"(c));
  return d;
#endif
}

__global__ __launch_bounds__(256) void upsample_attn_kernel(
    const float* __restrict__ x0g, const float* __restrict__ x1g,
    const float* __restrict__ x2g, const float* __restrict__ wg,
    float* __restrict__ outg)
{
  extern __shared__ float smem[];
  const int tid = threadIdx.x;
  const int b   = blockIdx.y;
  const int i0  = blockIdx.x << 2;          // first of 4 base rows

  // ---- phase 1a: zero halo / table region (x tiles + tbl) ----
  for (int idx = tid; idx < WT_OFF; idx += 256) smem[idx] = 0.0f;
  __syncthreads();

  // ---- phase 1b: fill LDS ----
  // x0 interior: 6 rows x 64 cols
  for (int idx = tid; idx < 6 * 64; idx += 256) {
    int rr = idx >> 6, c = idx & 63;
    int r = i0 - 1 + rr;
    if ((unsigned)r < 64u)
      smem[X0_OFF + rr * 66 + c + 1] = x0g[(size_t)b * 4096 + r * 64 + c];
  }
  // x1 interior: 12 rows x 128 cols
  for (int idx = tid; idx < 12 * 128; idx += 256) {
    int rr = idx >> 7, c = idx & 127;
    int r = 2 * i0 - 2 + rr;
    if ((unsigned)r < 128u)
      smem[X1_OFF + rr * 132 + c + 2] = x1g[(size_t)b * 16384 + r * 128 + c];
  }
  // x2 interior: 24 rows x 256 cols (row 24 left zero for K-padding)
  for (int idx = tid; idx < 24 * 256; idx += 256) {
    int rr = idx >> 8, c = idx & 255;
    int r = 4 * i0 - 4 + rr;
    if ((unsigned)r < 256u)
      smem[X2_OFF + rr * 264 + c + 4] = x2g[(size_t)b * 65536 + r * 256 + c];
  }
  // weight: element (k,n) -> WT_OFF + (k/2)*160 + 2n + (k&1); pad k>=189 -> 0
  for (int idx = tid; idx < 192 * 64; idx += 256) {
    int k = idx >> 6, n = idx & 63;
    smem[WT_OFF + (k >> 1) * WT_PITCH + (n << 1) + (k & 1)] =
        (k < 189) ? wg[k * 64 + n] : 0.0f;
  }
  // patch-offset table
  if (tid < 192) {
    int k = tid, e;
    if (k < 9)        { e = (k / 3) * 66 + (k % 3); }
    else if (k < 45)  { int kk = k - 9;  e = (kk / 6) * 132 + (kk % 6); }
    else              { int kk = k - 45; e = (kk / 12) * 264 + (kk % 12); }
    ((int*)smem)[TBL_OFF + k] = e;
  }
  __syncthreads();

  // ---- phase 2: WMMA k-loop ----
  const int lane = tid & 31;
  const int wv   = tid >> 5;
  const int m    = lane & 15;      // A: row M / B,C,D: col N (lane%16)
  const int lh   = lane >> 4;      // lane half -> K sub-block / M high half
  const int il   = wv >> 1;        // local base row 0..3
  const int jA   = ((wv & 1) << 5) + m;   // M-group 0 column
  const int jB   = jA + 16;               // M-group 1 column

  const int q0A = X0_OFF + il * 66   + jA;
  const int q0B = X0_OFF + il * 66   + jB;
  const int q1A = X1_OFF + il * 264  + (jA << 1);
  const int q1B = X1_OFF + il * 264  + (jB << 1);
  const int q2A = X2_OFF + il * 1056 + (jA << 2);
  const int q2B = X2_OFF + il * 1056 + (jB << 2);

  const int* tbl = ((const int*)smem) + TBL_OFF;
  const int  bwm = WT_OFF + (m << 1);     // + t*32 for N-tile t, + (ka>>1)*160

  v8f accA[4], accB[4];
#pragma unroll
  for (int t = 0; t < 4; ++t) {
    accA[t] = (v8f){0.f, 0.f, 0.f, 0.f, 0.f, 0.f, 0.f, 0.f};
    accB[t] = accA[t];
  }

#pragma unroll 2
  for (int s = 0; s < 48; ++s) {
    const int ka = (s << 2) + (lh << 1);          // this lane-half's K base
    const int2 ee = *(const int2*)(tbl + ka);     // patch offsets k, k+1
    const int s0 = (ka     < 9) ? 0 : ((ka     < 45) ? 1 : 2);
    const int s1 = (ka + 1 < 9) ? 0 : ((ka + 1 < 45) ? 1 : 2);
    const int qa0 = (s0 == 0) ? q0A : ((s0 == 1) ? q1A : q2A);
    const int qa1 = (s1 == 0) ? q0A : ((s1 == 1) ? q1A : q2A);
    const int qb0 = (s0 == 0) ? q0B : ((s0 == 1) ? q1B : q2B);
    const int qb1 = (s1 == 0) ? q0B : ((s1 == 1) ? q1B : q2B);

    const v2f A0 = { smem[qa0 + ee.x], smem[qa1 + ee.y] };
    const v2f A1 = { smem[qb0 + ee.x], smem[qb1 + ee.y] };

    const int wrow = bwm + (ka >> 1) * WT_PITCH;
#pragma unroll
    for (int t = 0; t < 4; ++t) {
      const float2 bb = *(const float2*)(smem + wrow + (t << 5));
      const v2f B = { bb.x, bb.y };
      accA[t] = wmma4(A0, B, accA[t]);
      accB[t] = wmma4(A1, B, accB[t]);
    }
  }

  // ---- phase 3: ReLU + spill logits to LDS (aliases x1/x2/Wt) ----
  __syncthreads();                         // everyone done reading tiles/W
  float* sLog = smem + LOG_OFF;
  const int baseA = wv << 5;               // positions 32*wv .. +15
  const int baseB = baseA + 16;
#pragma unroll
  for (int t = 0; t < 4; ++t) {
#pragma unroll
    for (int r = 0; r < 8; ++r) {
      const int pA = baseA + r + (lh << 3);   // D layout: M = r + 8*half
      const int pB = baseB + r + (lh << 3);
      sLog[pA * LOG_PITCH + (t << 4) + m] = fmaxf(accA[t][r], 0.0f);
      sLog[pB * LOG_PITCH + (t << 4) + m] = fmaxf(accB[t][r], 0.0f);
    }
  }
  __syncthreads();

  // ---- phase 4: softmax(64) + scale by x0 pixel + pixel-shuffle store ----
  const int p   = tid;                     // one position per thread
  const int ilp = p >> 6, jp = p & 63;
  float4* r4 = (float4*)(sLog + p * LOG_PITCH);

  float mx = -3.402823466e38f;
#pragma unroll
  for (int c = 0; c < 16; ++c) {
    const float4 v = r4[c];
    mx = fmaxf(mx, fmaxf(fmaxf(v.x, v.y), fmaxf(v.z, v.w)));
  }
  float sum = 0.0f;
#pragma unroll
  for (int c = 0; c < 16; ++c) {
    float4 v = r4[c];
    v.x = __expf(v.x - mx); v.y = __expf(v.y - mx);
    v.z = __expf(v.z - mx); v.w = __expf(v.w - mx);
    sum += v.x + v.y + v.z + v.w;
    r4[c] = v;
  }
  const float x0v = smem[X0_OFF + (ilp + 1) * 66 + jp + 1];
  const float sc  = x0v / sum;

  // out[b, 0, (i0+ilp)*8 + u, jp*8 + v], channel c = u*8+v
  const size_t ob = ((size_t)b * 512 + (size_t)(i0 + ilp) * 8) * 512
                  + (size_t)jp * 8;
#pragma unroll
  for (int u = 0; u < 8; ++u) {
    float4 lo = r4[2 * u], hi = r4[2 * u + 1];
    lo.x *= sc; lo.y *= sc; lo.z *= sc; lo.w *= sc;
    hi.x *= sc; hi.y *= sc; hi.z *= sc; hi.w *= sc;
    *(float4*)(outg + ob + (size_t)u * 512)     = lo;
    *(float4*)(outg + ob + (size_t)u * 512 + 4) = hi;
  }
}

extern "C" void kernel_launch(void* const* d_in, const int* in_sizes, int n_in,
                              void* d_out, int out_size, void* d_ws, size_t ws_size,
                              hipStream_t stream) {
  (void)in_sizes; (void)n_in; (void)out_size; (void)d_ws; (void)ws_size;
  const float* x0 = (const float*)d_in[0];
  const float* x1 = (const float*)d_in[1];
  const float* x2 = (const float*)d_in[2];
  const float* w  = (const float*)d_in[3];
  float* out = (float*)d_out;

  // allow > default dynamic LDS (96.5 KB); deterministic, not a stream op
  (void)hipFuncSetAttribute((const void*)upsample_attn_kernel,
                            hipFuncAttributeMaxDynamicSharedMemorySize,
                            SMEM_BYTES);

  dim3 grid(16, 64, 1);   // 16 row-quads x 64 batches
  upsample_attn_kernel<<<grid, 256, SMEM_BYTES, stream>>>(x0, x1, x2, w, out);
}